// GCN_35296041238721
// MI455X (gfx1250) — compile-verified
//
#include <hip/hip_runtime.h>
#include <math.h>

#define N_NODES 50000
#define IN_F    128
#define OUT_F   128
#define N_EDGES 500000

typedef __attribute__((ext_vector_type(2))) float v2f;
typedef __attribute__((ext_vector_type(8))) float v8f;

// ---------------------------------------------------------------------------
// 1) deg[v] = 1 (self loop)
__global__ void gcn_deg_init(float* __restrict__ deg) {
    int i = blockIdx.x * blockDim.x + threadIdx.x;
    if (i < N_NODES) deg[i] = 1.0f;
}

// 2) deg[col[e]] += 1
__global__ void gcn_deg_count(const int* __restrict__ col, float* __restrict__ deg) {
    int e = blockIdx.x * blockDim.x + threadIdx.x;
    if (e < N_EDGES) atomicAdd(&deg[col[e]], 1.0f);
}

// 3) deg -> deg^{-1/2} in place (deg >= 1 always, thanks to self loop)
__global__ void gcn_rsqrt(float* __restrict__ deg) {
    int i = blockIdx.x * blockDim.x + threadIdx.x;
    if (i < N_NODES) deg[i] = rsqrtf(deg[i]);
}

// ---------------------------------------------------------------------------
// 4) hs = (x @ W) * dinv[row]  via V_WMMA_F32_16X16X4_F32.
//    Also seeds out = hs (the self-loop contribution, pre-final-scale).
//    Block: 256 thr = 8 waves; block tile = 64 rows x 128 cols.
//    Wave tile: 16 rows x 64 cols (4 accumulators, A fragment reused 4x).
//    W staged in LDS as k-pair-major float2 so every B fragment is a single
//    contiguous ds_load_b64 into an even-aligned VGPR pair.
__launch_bounds__(256)
__global__ void gcn_gemm_wmma(const float* __restrict__ x,
                              const float* __restrict__ W,
                              const float* __restrict__ dinv,
                              float* __restrict__ hs,
                              float* __restrict__ out) {
    // Wp[(k>>1)*128 + n] = { W[k][n], W[k+1][n] }  -> 8192 float2 = 64 KB
    __shared__ v2f Wp[(IN_F / 2) * OUT_F];

    {
        float* lds = (float*)Wp;
        for (int i = threadIdx.x; i < IN_F * OUT_F; i += 256) {
            const int k = i >> 7;          // row of W (K index)
            const int n = i & 127;         // col of W (N index)
            lds[((k >> 1) << 8) + (n << 1) + (k & 1)] = W[i];
        }
    }
    __syncthreads();

    const int lane = threadIdx.x & 31;
    const int wave = threadIdx.x >> 5;
    const int mt   = wave >> 1;        // 0..3 : which 16-row subtile
    const int nh   = wave &  1;        // 0..1 : which 64-col half

    const int tile_row0 = blockIdx.x * 64 + mt * 16;   // wave-uniform
    if (tile_row0 >= N_NODES) return;  // whole-wave early out (50000 % 16 == 0)

    // A fragment (32-bit A 16x4): lane -> row = lane&15, Kpair = (lane>>4)*2
    const int kb   = (lane >> 4) * 2;
    const float* xrow = x + (size_t)(tile_row0 + (lane & 15)) * IN_F;

    const int nlo = lane & 15;                 // B/C fragment column within 16-tile
    const int n0  = nh * 64 + nlo;             // this lane's base output column
    const v2f* __restrict__ Wrow = Wp + n0;    // indexed by pair-k * 128

    v8f acc0 = {}, acc1 = {}, acc2 = {}, acc3 = {};

    #pragma unroll 4
    for (int k0 = 0; k0 < IN_F; k0 += 4) {
        const v2f a  = *(const v2f*)(xrow + k0 + kb);   // global_load_b64
        const int kp = (k0 + kb) >> 1;                   // k-pair index
        const v2f* __restrict__ wp = Wrow + (kp << 7);

        v2f b = wp[0];                                   // ds_load_b64
        acc0 = __builtin_amdgcn_wmma_f32_16x16x4_f32(false, a, false, b, (short)0, acc0, false, false);
        b = wp[16];
        acc1 = __builtin_amdgcn_wmma_f32_16x16x4_f32(false, a, false, b, (short)0, acc1, false, false);
        b = wp[32];
        acc2 = __builtin_amdgcn_wmma_f32_16x16x4_f32(false, a, false, b, (short)0, acc2, false, false);
        b = wp[48];
        acc3 = __builtin_amdgcn_wmma_f32_16x16x4_f32(false, a, false, b, (short)0, acc3, false, false);
    }

    // C/D fragment (16x16 f32): VGPR r holds row (lane<16 ? r : 8+r), col = lane&15.
    const int rbase = tile_row0 + ((lane >> 4) * 8);
    #pragma unroll
    for (int r = 0; r < 8; ++r) {
        const int row = rbase + r;
        const float s = dinv[row];
        float v0 = acc0[r] * s;
        float v1 = acc1[r] * s;
        float v2 = acc2[r] * s;
        float v3 = acc3[r] * s;
        size_t base = (size_t)row * OUT_F + n0;
        hs [base +  0] = v0;  out[base +  0] = v0;
        hs [base + 16] = v1;  out[base + 16] = v1;
        hs [base + 32] = v2;  out[base + 32] = v2;
        hs [base + 48] = v3;  out[base + 48] = v3;
    }
}

// ---------------------------------------------------------------------------
// 5) out[col[e]] += hs[row[e]]   — one wave (32 lanes) per edge, float4 per lane.
//    hs (25.6 MB) and out (25.6 MB) both live in the 192 MB L2 -> on-chip traffic.
__launch_bounds__(256)
__global__ void gcn_scatter(const int* __restrict__ ei,
                            const float* __restrict__ hs,
                            float* __restrict__ out) {
    const unsigned tid = blockIdx.x * 256u + threadIdx.x;
    const unsigned e   = tid >> 5;
    if (e >= N_EDGES) return;
    const int f = (tid & 31) * 4;
    const int r = ei[e];             // edge_index[0][e] : source
    const int c = ei[N_EDGES + e];   // edge_index[1][e] : target
    const float4 v = *(const float4*)(hs + (size_t)r * OUT_F + f);
    float* o = out + (size_t)c * OUT_F + f;
    atomicAdd(o + 0, v.x);
    atomicAdd(o + 1, v.y);
    atomicAdd(o + 2, v.z);
    atomicAdd(o + 3, v.w);
}

// ---------------------------------------------------------------------------
// 6) out[c][f] = out[c][f] * dinv[c] + b[f]
__global__ void gcn_finalize(const float* __restrict__ dinv,
                             const float* __restrict__ bias,
                             float* __restrict__ out) {
    unsigned i = blockIdx.x * blockDim.x + threadIdx.x;
    if (i < (unsigned)N_NODES * OUT_F)
        out[i] = out[i] * dinv[i >> 7] + bias[i & (OUT_F - 1)];
}

// ---------------------------------------------------------------------------
extern "C" void kernel_launch(void* const* d_in, const int* in_sizes, int n_in,
                              void* d_out, int out_size, void* d_ws, size_t ws_size,
                              hipStream_t stream) {
    (void)in_sizes; (void)n_in; (void)out_size; (void)ws_size;

    const float* x    = (const float*)d_in[0];
    const int*   ei   = (const int*)d_in[1];   // [2, N_EDGES], row-major int32
    const float* W    = (const float*)d_in[2];
    const float* bias = (const float*)d_in[3];
    float*       out  = (float*)d_out;

    // Workspace layout: [0, N) deg->dinv ; then N*128 floats of hs (16B aligned:
    // 50000*4 = 200000 bytes, multiple of 16).
    float* deg = (float*)d_ws;
    float* hs  = (float*)d_ws + N_NODES;

    gcn_deg_init <<<(N_NODES + 255) / 256, 256, 0, stream>>>(deg);
    gcn_deg_count<<<(N_EDGES + 255) / 256, 256, 0, stream>>>(ei + N_EDGES, deg);
    gcn_rsqrt    <<<(N_NODES + 255) / 256, 256, 0, stream>>>(deg);

    // 782 blocks * 64 rows = 50048 >= 50000
    gcn_gemm_wmma<<<(N_NODES + 63) / 64, 256, 0, stream>>>(x, W, deg, hs, out);

    // 32 lanes per edge
    gcn_scatter<<<(N_EDGES * 32) / 256, 256, 0, stream>>>(ei, hs, out);

    gcn_finalize<<<((unsigned)N_NODES * OUT_F + 255) / 256, 256, 0, stream>>>(deg, bias, out);
}